// SyncMaskedBatchNorm1d_30889404792984
// MI455X (gfx1250) — compile-verified
//
#include <hip/hip_runtime.h>
#include <stdint.h>
#include <stddef.h>

typedef __attribute__((ext_vector_type(2))) float v2f;
typedef __attribute__((ext_vector_type(8))) float v8f;

#define C_CH   512
#define L_LEN  4096
#define B_SZ   32
#define NSLICE 16
#define BN_EPS 1e-5f

// ---------------------------------------------------------------------------
// Stage 1: per-(channel, slice) partial sums of x*m and (x*m)^2.
// grid = (C_CH, NSLICE), block = 256. Slice s covers batches {2s, 2s+1}.
// Channel-0 blocks also integer-count the mask (deterministic int atomics).
// ---------------------------------------------------------------------------
__global__ void __launch_bounds__(256)
stats_kernel(const float* __restrict__ x, const int* __restrict__ mask,
             float* __restrict__ psum, float* __restrict__ psq,
             int* __restrict__ cnt) {
    __shared__ float rs[256];
    __shared__ float rq[256];
    __shared__ int   ri[256];

    const int c   = blockIdx.x;
    const int s   = blockIdx.y;
    const int tid = threadIdx.x;
    const bool count_mask = (c == 0);

    float sum = 0.0f, sq = 0.0f;
    int   mc  = 0;

#pragma unroll
    for (int i = 0; i < 8; ++i) {
        const int idx = i * 256 + tid;      // 0..2047 float4 groups
        const int bo  = idx >> 10;          // 0..1
        const int l4  = idx & 1023;         // float4 index within row
        const int b   = s * 2 + bo;
        const float4 xv = *reinterpret_cast<const float4*>(
            x + ((size_t)b * C_CH + c) * L_LEN + (size_t)l4 * 4);
        const int4 mv = *reinterpret_cast<const int4*>(
            mask + (size_t)b * L_LEN + (size_t)l4 * 4);
        const float t0 = xv.x * (float)mv.x;
        const float t1 = xv.y * (float)mv.y;
        const float t2 = xv.z * (float)mv.z;
        const float t3 = xv.w * (float)mv.w;
        sum += t0; sum += t1; sum += t2; sum += t3;
        sq = fmaf(t0, t0, sq); sq = fmaf(t1, t1, sq);
        sq = fmaf(t2, t2, sq); sq = fmaf(t3, t3, sq);
        if (count_mask) mc += mv.x + mv.y + mv.z + mv.w;
    }

    rs[tid] = sum; rq[tid] = sq; ri[tid] = mc;
    __syncthreads();
    for (int off = 128; off > 0; off >>= 1) {
        if (tid < off) {
            rs[tid] += rs[tid + off];
            rq[tid] += rq[tid + off];
            ri[tid] += ri[tid + off];
        }
        __syncthreads();
    }
    if (tid == 0) {
        psum[s * C_CH + c] = rs[0];
        psq [s * C_CH + c] = rq[0];
        if (count_mask) atomicAdd(cnt, ri[0]);
    }
}

// ---------------------------------------------------------------------------
// Stage 2: WMMA cross-slice reduction + coefficient computation.
// One wave per 16-channel tile: A = ones(16x4), B tiles carry 4 slices of
// partials per call; with A==1 every D element is the 16-slice column sum,
// independent of the f32 B-matrix K-slot permutation (N = lane&15 only).
// grid = C_CH/16, block = 32 (full wave, EXEC all ones through the WMMAs).
// ---------------------------------------------------------------------------
__global__ void __launch_bounds__(32)
finalize_kernel(const float* __restrict__ psum, const float* __restrict__ psq,
                const int* __restrict__ cnt,
                const float* __restrict__ weight, const float* __restrict__ bias,
                float* __restrict__ Acoef, float* __restrict__ Bcoef) {
    const int lane = threadIdx.x;
    const int n    = lane & 15;
    const int half = lane >> 4;
    const int c    = blockIdx.x * 16 + n;

    v2f ones; ones.x = 1.0f; ones.y = 1.0f;
    v8f acc  = {};
    v8f accq = {};

#pragma unroll
    for (int j = 0; j < 4; ++j) {
        const int s0 = 4 * j + half;        // distinct slice per {half, vgpr} slot
        const int s1 = 4 * j + 2 + half;
        v2f bs; bs.x = psum[s0 * C_CH + c]; bs.y = psum[s1 * C_CH + c];
        v2f bq; bq.x = psq [s0 * C_CH + c]; bq.y = psq [s1 * C_CH + c];
        acc  = __builtin_amdgcn_wmma_f32_16x16x4_f32(false, ones, false, bs,
                                                     (short)0, acc,  false, false);
        accq = __builtin_amdgcn_wmma_f32_16x16x4_f32(false, ones, false, bq,
                                                     (short)0, accq, false, false);
    }

    const float stot = acc[0];    // every row of D equals the column sum
    const float qtot = accq[0];
    const float cntf = fmaxf((float)(*cnt), 1.0f);
    const float mean = stot / cntf;
    const float var  = fmaxf(qtot / cntf - mean * mean, 0.0f);
    const float inv  = rsqrtf(var + BN_EPS);
    const float a    = inv * weight[c];
    const float b0   = bias[c] - mean * a;
    if (lane < 16) {              // lanes 16..31 duplicate channels
        Acoef[c] = a;
        Bcoef[c] = b0;
    }
}

// ---------------------------------------------------------------------------
// Stage 3: normalize. Block = (b, 8-channel tile, 1024-element l chunk).
// Mask chunk (4KB) staged to LDS with global_load_async_to_lds_b128, read once
// into registers, reused across all 8 channels. out = m ? fma(x,a,b0) : bias.
// grid = (L/1024, C/8, B), block = 256.
// ---------------------------------------------------------------------------
__global__ void __launch_bounds__(256)
norm_kernel(const float* __restrict__ x, const int* __restrict__ mask,
            const float* __restrict__ Acoef, const float* __restrict__ Bcoef,
            const float* __restrict__ bias, float* __restrict__ out) {
    __shared__ int smask[1024];

    const int tid = threadIdx.x;
    const int b   = blockIdx.z;
    const int c0  = blockIdx.y * 8;
    const int l0  = blockIdx.x * 1024;

    // Async copy: 256 threads x 16B = 4KB of mask into LDS.
    {
        const unsigned lds_addr =
            (unsigned)(uintptr_t)(&smask[tid * 4]);        // low 32b of flat = LDS offset
        const unsigned long long gaddr =
            (unsigned long long)(uintptr_t)(mask + (size_t)b * L_LEN + l0 + tid * 4);
        asm volatile("global_load_async_to_lds_b128 %0, %1, off"
                     :: "v"(lds_addr), "v"(gaddr) : "memory");
        asm volatile("s_wait_asynccnt 0" ::: "memory");
    }
    __syncthreads();

    const int4 m4 = *reinterpret_cast<const int4*>(&smask[tid * 4]);
    const size_t xbase = ((size_t)b * C_CH + c0) * L_LEN + (size_t)l0 + (size_t)tid * 4;

#pragma unroll
    for (int cc = 0; cc < 8; ++cc) {
        const int   c  = c0 + cc;
        const float a  = Acoef[c];
        const float b0 = Bcoef[c];
        const float bi = bias[c];
        const float4 xv = *reinterpret_cast<const float4*>(x + xbase + (size_t)cc * L_LEN);
        float4 o;
        o.x = m4.x ? fmaf(xv.x, a, b0) : bi;
        o.y = m4.y ? fmaf(xv.y, a, b0) : bi;
        o.z = m4.z ? fmaf(xv.z, a, b0) : bi;
        o.w = m4.w ? fmaf(xv.w, a, b0) : bi;
        *reinterpret_cast<float4*>(out + xbase + (size_t)cc * L_LEN) = o;
    }
}

// ---------------------------------------------------------------------------
// Workspace layout (floats): [0]=int mask count | +64: psum[16*512] |
// psq[16*512] | Acoef[512] | Bcoef[512]   (~68 KB total)
// ---------------------------------------------------------------------------
extern "C" void kernel_launch(void* const* d_in, const int* in_sizes, int n_in,
                              void* d_out, int out_size, void* d_ws, size_t ws_size,
                              hipStream_t stream) {
    const float* x      = (const float*)d_in[0];
    const int*   mask   = (const int*)d_in[1];
    const float* weight = (const float*)d_in[2];
    const float* bias   = (const float*)d_in[3];
    float*       out    = (float*)d_out;

    float* wsf  = (float*)d_ws;
    int*   cnt  = (int*)d_ws;
    float* psum = wsf + 64;
    float* psq  = psum + NSLICE * C_CH;
    float* Ac   = psq  + NSLICE * C_CH;
    float* Bc   = Ac   + C_CH;

    hipMemsetAsync(cnt, 0, sizeof(int), stream);

    stats_kernel<<<dim3(C_CH, NSLICE), 256, 0, stream>>>(x, mask, psum, psq, cnt);
    finalize_kernel<<<C_CH / 16, 32, 0, stream>>>(psum, psq, cnt, weight, bias, Ac, Bc);
    norm_kernel<<<dim3(L_LEN / 1024, C_CH / 8, B_SZ), 256, 0, stream>>>(
        x, mask, Ac, Bc, bias, out);
}